// MaskedPatchEmbedding_22754736734866
// MI455X (gfx1250) — compile-verified
//
#include <hip/hip_runtime.h>
#include <hip/hip_bf16.h>

typedef float v2f __attribute__((ext_vector_type(2)));
typedef float v8f __attribute__((ext_vector_type(8)));

#define BATCH   128
#define E_DIM   768
#define K_DIM   768     // 3*16*16
#define NPATCH  196     // 14*14
#define SEQ     197
#define NMASK   75

#define M_TILE  128
#define N_TILE  64
#define K_TILE  32
#define LDSS    36      // padded LDS row stride (floats): 16B aligned, bank-conflict free

// ---------------------------------------------------------------------------
// GEMM: out[b, 1+p, e] = sum_k im2col(src)[b*196+p, k] * proj[e, k]
// ---------------------------------------------------------------------------
__global__ __launch_bounds__(256) void mpe_patch_gemm(
    const float* __restrict__ src,    // [128,3,224,224]
    const float* __restrict__ proj,   // [768,768]  (e-major, k contiguous)
    float* __restrict__ out)          // [128,197,768]
{
    __shared__ float lds_a[M_TILE * LDSS];   // 18.4 KB
    __shared__ float lds_b[N_TILE * LDSS];   //  9.2 KB

    const int tid = threadIdx.x;
    const int m0  = blockIdx.y * M_TILE;     // 196 tiles over M = 25088
    const int n0  = blockIdx.x * N_TILE;     // 12 tiles over N = 768

    // ---- A-tile loading assignment: 2 threads/row, 1 segment (16 floats) each
    const int arow = tid & 127;
    const int aseg = tid >> 7;                        // which of 2 h-rows in this k-slice
    const int am   = m0 + arow;
    const int ab   = am / NPATCH;
    const int ap   = am - ab * NPATCH;
    const int aph  = ap / 14;
    const int apw  = ap - aph * 14;
    const int arowbase = ((ab * 3) * 224 + aph * 16) * 224 + apw * 16;

    // ---- B-tile loading assignment: 8 threads/row (32 floats), 2 rows/thread
    const int bq      = tid & 7;
    const int bn_base = tid >> 3;                     // + j*32

    // ---- WMMA fragment assignment
    const int lane  = tid & 31;
    const int wv    = tid >> 5;                       // 8 waves
    const int wm    = (wv & 3) * 32;                  // wave M offset (4 rows of waves)
    const int wn    = (wv >> 2) * 32;                 // wave N offset (2 cols of waves)
    const int lrow  = lane & 15;
    const int khalf = lane >> 4;                      // A lanes 0-15: K=0,1 | 16-31: K=2,3

    v8f acc00 = {}; v8f acc01 = {}; v8f acc10 = {}; v8f acc11 = {};

    float4 va[4];
    float4 vb[2];

    // prologue: stage k0 = 0
    {
        const float* s = src + arowbase + aseg * 224; // c=0, h0=0
        #pragma unroll
        for (int q = 0; q < 4; ++q) va[q] = *(const float4*)(s + q * 4);
        #pragma unroll
        for (int j = 0; j < 2; ++j)
            vb[j] = *(const float4*)(proj + (n0 + bn_base + j * 32) * K_DIM + bq * 4);
    }

    for (int k0 = 0; k0 < K_DIM; k0 += K_TILE) {
        __syncthreads();
        #pragma unroll
        for (int q = 0; q < 4; ++q)
            *(float4*)&lds_a[arow * LDSS + aseg * 16 + q * 4] = va[q];
        #pragma unroll
        for (int j = 0; j < 2; ++j)
            *(float4*)&lds_b[(bn_base + j * 32) * LDSS + bq * 4] = vb[j];
        __syncthreads();

        // prefetch next K-tile into registers while WMMAs run
        const int kn = k0 + K_TILE;
        if (kn < K_DIM) {
            const int c  = kn >> 8;                   // channel (256 k per channel)
            const int h0 = (kn & 255) >> 4;           // h-row pair inside patch
            const float* s = src + arowbase + (c * 224 + h0 + aseg) * 224;
            #pragma unroll
            for (int q = 0; q < 4; ++q) va[q] = *(const float4*)(s + q * 4);
            #pragma unroll
            for (int j = 0; j < 2; ++j)
                vb[j] = *(const float4*)(proj + (n0 + bn_base + j * 32) * K_DIM + kn + bq * 4);
        }

        #pragma unroll
        for (int kt = 0; kt < 8; ++kt) {
            const int ka = kt * 4 + khalf * 2;
            v2f a0 = *(const v2f*)&lds_a[(wm +      lrow) * LDSS + ka];
            v2f a1 = *(const v2f*)&lds_a[(wm + 16 + lrow) * LDSS + ka];
            v2f b0 = *(const v2f*)&lds_b[(wn +      lrow) * LDSS + ka];
            v2f b1 = *(const v2f*)&lds_b[(wn + 16 + lrow) * LDSS + ka];
            acc00 = __builtin_amdgcn_wmma_f32_16x16x4_f32(false, a0, false, b0, (short)0, acc00, false, false);
            acc01 = __builtin_amdgcn_wmma_f32_16x16x4_f32(false, a0, false, b1, (short)0, acc01, false, false);
            acc10 = __builtin_amdgcn_wmma_f32_16x16x4_f32(false, a1, false, b0, (short)0, acc10, false, false);
            acc11 = __builtin_amdgcn_wmma_f32_16x16x4_f32(false, a1, false, b1, (short)0, acc11, false, false);
        }
    }

    // epilogue: C/D layout -> VGPR v holds M = v + 8*khalf, N = lrow (per 16x16 tile)
    #pragma unroll
    for (int i = 0; i < 2; ++i) {
        const int   mbase = m0 + wm + i * 16 + khalf * 8;
        const v8f r0 = (i == 0) ? acc00 : acc10;
        const v8f r1 = (i == 0) ? acc01 : acc11;
        #pragma unroll
        for (int v = 0; v < 8; ++v) {
            const int mm = mbase + v;
            const int bb = mm / NPATCH;
            const int ll = 1 + (mm - bb * NPATCH);        // patch tokens start at l=1
            float* orow = out + (size_t)(bb * SEQ + ll) * E_DIM + n0 + wn + lrow;
            orow[0]  = r0[v];
            orow[16] = r1[v];
        }
    }
}

// ---------------------------------------------------------------------------
// cls token: out[b, 0, :] = cls_emb
// ---------------------------------------------------------------------------
__global__ __launch_bounds__(256) void mpe_fill_cls(
    const float* __restrict__ cls_emb, float* __restrict__ out)
{
    const int idx = blockIdx.x * 256 + threadIdx.x;   // [0, 128*768)
    const int b = idx / E_DIM;
    const int e = idx - b * E_DIM;
    out[(size_t)b * SEQ * E_DIM + e] = cls_emb[e];
}

// ---------------------------------------------------------------------------
// masked scatter: out[b, mask[b,j], :] = mask_emb  (duplicates benign)
// ---------------------------------------------------------------------------
__global__ __launch_bounds__(256) void mpe_scatter_mask(
    const int* __restrict__ mask,       // [128,75]
    const float* __restrict__ mask_emb, // [768]
    float* __restrict__ out)
{
    const int pair = blockIdx.x;                      // b*75 + j
    const int b = pair / NMASK;
    const int l = mask[pair];
    float* dst = out + (size_t)(b * SEQ + l) * E_DIM;
    for (int e = threadIdx.x; e < E_DIM; e += 256)
        dst[e] = mask_emb[e];
}

extern "C" void kernel_launch(void* const* d_in, const int* in_sizes, int n_in,
                              void* d_out, int out_size, void* d_ws, size_t ws_size,
                              hipStream_t stream) {
    (void)in_sizes; (void)n_in; (void)out_size; (void)d_ws; (void)ws_size;
    const float* src      = (const float*)d_in[0];
    const int*   mask     = (const int*)d_in[1];
    const float* proj_w   = (const float*)d_in[2];
    const float* cls_emb  = (const float*)d_in[3];
    const float* mask_emb = (const float*)d_in[4];
    float*       out      = (float*)d_out;

    dim3 gg(E_DIM / N_TILE, (BATCH * NPATCH) / M_TILE);   // 12 x 196
    mpe_patch_gemm<<<gg, 256, 0, stream>>>(src, proj_w, out);
    mpe_fill_cls<<<(BATCH * E_DIM) / 256, 256, 0, stream>>>(cls_emb, out);
    mpe_scatter_mask<<<BATCH * NMASK, 256, 0, stream>>>(mask, mask_emb, out);
}